// GAT_Net_21801253994541
// MI455X (gfx1250) — compile-verified
//
#include <hip/hip_runtime.h>
#include <math.h>

// ---------------------------------------------------------------------------
// GAT 4-layer pipeline for MI455X (gfx1250, wave32).
// GEMMs use V_WMMA_F32_16X16X4_F32 (full fp32 fidelity; GEMM cost ~7 GFLOP is
// trivial vs. the ~1.7GB memory-bound edge-aggregation phase, so no precision
// drop needed). K/M are template constants so the WMMA inner loop is pure
// {global_load_b64 A, 2x global_load_b32 B, v_wmma} with immediate addressing
// and no exec-mask round-trips. Edge softmax = 3 atomic passes; the scatter
// target (51MB) is L2-resident (192MB).
// ---------------------------------------------------------------------------

typedef __attribute__((ext_vector_type(2))) float v2f;
typedef __attribute__((ext_vector_type(8))) float v8f;

#define NEG_SLOPE 0.2f

// ---------------- fp32 WMMA GEMM: C[N,M] = A[N,K] @ B[K,M] -----------------
// One wave per 16x16 output tile, stepping K by 4.
// A (16x4 f32): lanes 0-15 -> rows; VGPR0/1 hold K{0,1} (lanes<16) or K{2,3}
// (lanes>=16) => one contiguous 64-bit load per lane per step.
// B (4x16): row-striped across lanes within a VGPR, same half-split pattern.
// GUARD handles ragged M (=40): loads are clamped+zeroed branchlessly.
template <int K, int M, bool GUARD>
__global__ __launch_bounds__(32) void gemm_wmma_f32(
    const float* __restrict__ A, const float* __restrict__ B,
    float* __restrict__ C) {
  const int lane = threadIdx.x;
  const int half = lane >> 4;           // 0: lanes 0-15, 1: lanes 16-31
  const int l16  = lane & 15;
  const int row0 = blockIdx.y << 4;
  const int col  = (blockIdx.x << 4) + l16;
  const bool colok = GUARD ? (col < M) : true;
  const int colc = GUARD ? (colok ? col : M - 1) : col;   // clamped (safe) addr

  const float* __restrict__ arow = A + (size_t)(row0 + l16) * K + 2 * half;
  const float* __restrict__ bp   = B + (size_t)(2 * half) * M + colc;
  v8f acc = {0.f, 0.f, 0.f, 0.f, 0.f, 0.f, 0.f, 0.f};

#pragma unroll
  for (int k = 0; k < K; k += 4) {
    v2f a, b;
    a.x = arow[k];
    a.y = arow[k + 1];
    float bx = bp[(size_t)k * M];
    float by = bp[(size_t)k * M + M];
    if (GUARD && !colok) { bx = 0.f; by = 0.f; }          // v_cndmask, no exec
    b.x = bx;
    b.y = by;
    acc = __builtin_amdgcn_wmma_f32_16x16x4_f32(
        /*neg_a=*/false, a, /*neg_b=*/false, b,
        /*c_mod=*/(short)0, acc, /*reuse_a=*/false, /*reuse_b=*/false);
  }

  if (colok) {
#pragma unroll
    for (int j = 0; j < 8; ++j)
      C[(size_t)(row0 + 8 * half + j) * M + col] = acc[j];
  }
}

// ------------- attention logits: as[n,h] = <xw[n,h,:], a_src[h,:]> ----------
__global__ void attn_alpha(const float* __restrict__ xw,
                           const float* __restrict__ a_src,
                           const float* __restrict__ a_dst,
                           float* __restrict__ as, float* __restrict__ ad,
                           int N, int H, int C) {
  int i = blockIdx.x * blockDim.x + threadIdx.x;
  if (i >= N * H) return;
  int n = i / H, h = i - n * H;
  const float* __restrict__ row = xw + (size_t)n * H * C + (size_t)h * C;
  const float* __restrict__ vs = a_src + h * C;
  const float* __restrict__ vd = a_dst + h * C;
  float ss = 0.f, sd = 0.f;
  for (int c = 0; c < C; ++c) {
    float v = row[c];
    ss += v * vs[c];
    sd += v * vd[c];
  }
  as[i] = ss;
  ad[i] = sd;
}

__global__ void fill_f32(float* __restrict__ p, float v, long long n) {
  long long i = (long long)blockIdx.x * blockDim.x + threadIdx.x;
  if (i < n) p[i] = v;
}

__device__ __forceinline__ void edge_sd(const int* __restrict__ ei, int E,
                                        int e, int& s, int& d) {
  if (e < E) {
    s = ei[e];
    d = ei[E + e];
  } else {  // appended self-loop
    s = d = e - E;
  }
}

// ---------------- pass 1: segment max of leaky_relu(as[s]+ad[d]) -----------
__global__ void edge_max(const int* __restrict__ ei, int E, int EN, int H,
                         const float* __restrict__ as,
                         const float* __restrict__ ad, float* __restrict__ m) {
  long long i = (long long)blockIdx.x * blockDim.x + threadIdx.x;
  if (i >= (long long)EN * H) return;
  int e = (int)(i / H), h = (int)(i - (long long)e * H);
  int s, d;
  edge_sd(ei, E, e, s, d);
  float v = as[s * H + h] + ad[d * H + h];
  v = (v > 0.f) ? v : NEG_SLOPE * v;
  atomicMax(&m[d * H + h], v);  // global_atomic_max_num_f32
}

// ---------------- pass 2: p = exp(e - m[dst]); denom[dst] += p -------------
__global__ void edge_expsum(const int* __restrict__ ei, int E, int EN, int H,
                            const float* __restrict__ as,
                            const float* __restrict__ ad,
                            const float* __restrict__ m,
                            float* __restrict__ p, float* __restrict__ denom) {
  long long i = (long long)blockIdx.x * blockDim.x + threadIdx.x;
  if (i >= (long long)EN * H) return;
  int e = (int)(i / H), h = (int)(i - (long long)e * H);
  int s, d;
  edge_sd(ei, E, e, s, d);
  float v = as[s * H + h] + ad[d * H + h];
  v = (v > 0.f) ? v : NEG_SLOPE * v;
  float pe = expf(v - m[d * H + h]);
  p[i] = pe;
  atomicAdd(&denom[d * H + h], pe);
}

// ------- pass 3: out[dst,h,c] += xw[src,h,c] * p[e,h]/denom[dst,h] ---------
// Thread per (edge, h, c): consecutive threads touch consecutive columns of
// one edge's source row -> fully coalesced loads and atomics (L2-resident).
__global__ void edge_aggregate(const int* __restrict__ ei, int E, int EN,
                               int H, int C, const float* __restrict__ xw,
                               const float* __restrict__ p,
                               const float* __restrict__ denom,
                               float* __restrict__ out) {
  long long i = (long long)blockIdx.x * blockDim.x + threadIdx.x;
  int hc = H * C;
  if (i >= (long long)EN * hc) return;
  int e = (int)(i / hc);
  int rem = (int)(i - (long long)e * hc);
  int h = rem / C;
  int s, d;
  edge_sd(ei, E, e, s, d);
  float alpha = p[(long long)e * H + h] / denom[d * H + h];
  atomicAdd(&out[(size_t)d * hc + rem], xw[(size_t)s * hc + rem] * alpha);
}

// ---------------- epilogue: out += bias; optional ELU ----------------------
__global__ void bias_act(float* __restrict__ out, const float* __restrict__ b,
                         long long total, int HC, int apply_elu) {
  long long i = (long long)blockIdx.x * blockDim.x + threadIdx.x;
  if (i >= total) return;
  float v = out[i] + b[(int)(i % HC)];
  if (apply_elu) v = (v > 0.f) ? v : (expf(v) - 1.f);
  out[i] = v;
}

// ---------------- final row-wise log_softmax over C classes ----------------
__global__ void log_softmax_rows(const float* __restrict__ in,
                                 float* __restrict__ out, int N, int C) {
  int n = blockIdx.x * blockDim.x + threadIdx.x;
  if (n >= N) return;
  const float* __restrict__ row = in + (size_t)n * C;
  float mx = -INFINITY;
  for (int c = 0; c < C; ++c) mx = fmaxf(mx, row[c]);
  float s = 0.f;
  for (int c = 0; c < C; ++c) s += expf(row[c] - mx);
  float ls = logf(s);
  float* __restrict__ orow = out + (size_t)n * C;
  for (int c = 0; c < C; ++c) orow[c] = row[c] - mx - ls;
}

// ---------------------------------------------------------------------------
// Non-GEMM portion of one GAT layer (softmax-aggregate + epilogue).
static void run_gat_edges(const float* xw, const float* asrc, const float* adst,
                          const float* bias, int H, int C, const int* ei,
                          int E, int N, float* as, float* ad, float* m,
                          float* denom, float* p, float* out, int apply_elu,
                          hipStream_t stream) {
  const int M = H * C;
  const int EN = E + N;

  int nh = N * H;
  attn_alpha<<<(nh + 255) / 256, 256, 0, stream>>>(xw, asrc, adst, as, ad, N, H, C);

  fill_f32<<<(nh + 255) / 256, 256, 0, stream>>>(m, -INFINITY, nh);
  fill_f32<<<(nh + 255) / 256, 256, 0, stream>>>(denom, 0.f, nh);
  long long nm = (long long)N * M;
  fill_f32<<<(unsigned)((nm + 255) / 256), 256, 0, stream>>>(out, 0.f, nm);

  long long enh = (long long)EN * H;
  edge_max<<<(unsigned)((enh + 255) / 256), 256, 0, stream>>>(ei, E, EN, H, as, ad, m);
  edge_expsum<<<(unsigned)((enh + 255) / 256), 256, 0, stream>>>(ei, E, EN, H, as, ad, m, p, denom);

  long long tot = (long long)EN * M;
  edge_aggregate<<<(unsigned)((tot + 255) / 256), 256, 0, stream>>>(ei, E, EN, H, C, xw, p, denom, out);

  bias_act<<<(unsigned)((nm + 255) / 256), 256, 0, stream>>>(out, bias, nm, M, apply_elu);
}

extern "C" void kernel_launch(void* const* d_in, const int* in_sizes, int n_in,
                              void* d_out, int out_size, void* d_ws, size_t ws_size,
                              hipStream_t stream) {
  (void)n_in; (void)out_size; (void)ws_size;

  const int NFEAT = 256, H1 = 8, C1 = 32, NCLASS = 40;
  const int N = in_sizes[0] / NFEAT;   // 50000 (= 3125 * 16)
  const int E = in_sizes[1] / 2;       // 800000

  const float* x     = (const float*)d_in[0];
  const int*   ei    = (const int*)d_in[1];   // [2,E] int32 row-major
  const float* W1    = (const float*)d_in[2];
  const float* a1s   = (const float*)d_in[3];
  const float* a1d   = (const float*)d_in[4];
  const float* b1    = (const float*)d_in[5];
  const float* W2    = (const float*)d_in[6];
  const float* a2s   = (const float*)d_in[7];
  const float* a2d   = (const float*)d_in[8];
  const float* b2    = (const float*)d_in[9];
  const float* W3    = (const float*)d_in[10];
  const float* a3s   = (const float*)d_in[11];
  const float* a3d   = (const float*)d_in[12];
  const float* b3    = (const float*)d_in[13];

  // ---- workspace layout (floats) ----
  float* ws = (float*)d_ws;
  size_t off = 0;
  float* xw    = ws + off; off += (size_t)N * 256;       // GEMM output scratch
  float* h1    = ws + off; off += (size_t)N * 256;       // layer-1 output
  float* hA    = ws + off; off += (size_t)N * NCLASS;    // layer-2 / layer-4 out
  float* hB    = ws + off; off += (size_t)N * NCLASS;    // layer-3 output
  float* as    = ws + off; off += (size_t)N * H1;
  float* ad    = ws + off; off += (size_t)N * H1;
  float* m     = ws + off; off += (size_t)N * H1;
  float* denom = ws + off; off += (size_t)N * H1;
  float* p     = ws + off; off += (size_t)(E + N) * H1;

  const unsigned rowBlocks = (unsigned)(N / 16);  // 3125

  // Layer 1: [N,256]@[256,256] -> [N,8,32] concat -> [N,256], ELU
  gemm_wmma_f32<256, 256, false>
      <<<dim3(16, rowBlocks), 32, 0, stream>>>(x, W1, xw);
  run_gat_edges(xw, a1s, a1d, b1, H1, C1, ei, E, N, as, ad, m, denom, p, h1, 1, stream);

  // Layer 2: [N,256]@[256,40] -> [N,40], ELU
  gemm_wmma_f32<256, 40, true>
      <<<dim3(3, rowBlocks), 32, 0, stream>>>(h1, W2, xw);
  run_gat_edges(xw, a2s, a2d, b2, 1, NCLASS, ei, E, N, as, ad, m, denom, p, hA, 1, stream);

  // Layer 3: [N,40]@[40,40] -> [N,40], ELU (conv3)
  gemm_wmma_f32<40, 40, true>
      <<<dim3(3, rowBlocks), 32, 0, stream>>>(hA, W3, xw);
  run_gat_edges(xw, a3s, a3d, b3, 1, NCLASS, ei, E, N, as, ad, m, denom, p, hB, 1, stream);

  // Layer 4: conv3 reused, no ELU
  gemm_wmma_f32<40, 40, true>
      <<<dim3(3, rowBlocks), 32, 0, stream>>>(hB, W3, xw);
  run_gat_edges(xw, a3s, a3d, b3, 1, NCLASS, ei, E, N, as, ad, m, denom, p, hA, 0, stream);

  log_softmax_rows<<<(N + 255) / 256, 256, 0, stream>>>(hA, (float*)d_out, N, NCLASS);
}